// Emphasize_67078799229558
// MI455X (gfx1250) — compile-verified
//
#include <hip/hip_runtime.h>
#include <math.h>

// ---------------------------------------------------------------------------
// Emphasize: 2D FFT -> shifted-mask multiply -> 2D IFFT -> |.| on MI455X.
// FFT-256 four-step = two 16x16x16 complex matmuls on V_WMMA_F32_16X16X4_F32.
// fftshift/ifftshift (all axes incl. batch) folded into mask gather indices.
// Pass 2 tile movement uses GLOBAL_LOAD_ASYNC_TO_LDS_B128 /
// GLOBAL_STORE_ASYNC_FROM_LDS_B128 (ASYNCcnt).  Final output stores are
// non-temporal so the 67MB complex workspace stays resident in the 192MB L2.
// Needs ws_size >= B*256*256*8 bytes (B=128 -> 64 MB).
// ---------------------------------------------------------------------------

typedef float v2f __attribute__((ext_vector_type(2)));
typedef float v4f __attribute__((ext_vector_type(4)));
typedef float v8f __attribute__((ext_vector_type(8)));

#define WMMA_F32X4(A, B, C) \
  __builtin_amdgcn_wmma_f32_16x16x4_f32(false, (A), false, (B), (short)0, (C), false, false)

#define TWO_PI_F 6.283185307179586f

// LDS byte address of a __shared__ object: the LDS flat aperture keeps the
// offset in addr[31:0] (CDNA5 ISA 10.2), so truncation of the generic pointer
// yields the address the async-to-LDS VDST/VSRC VGPR expects.
__device__ __forceinline__ unsigned lds_addr_of(const void* p) {
  return (unsigned)(unsigned long long)(uintptr_t)p;
}

// Constant 16-point DFT matrix F[M][K] = exp(-2*pi*i*M*K/16) in the WMMA
// A-operand layout: chunk c covers K=4c..4c+3; lane L holds M=L%16,
// K = 4c + r + 2*(L/16) in v2f component r.
struct F16c {
  v2f fr[4];   // Re
  v2f fi[4];   // Im (forward)  = -sin
  v2f fin[4];  // -Im           = +sin (inverse Im, and -Fi terms)
};

__device__ __forceinline__ void build_F16(F16c& F, int lane) {
  const int M = lane & 15;
  const int hi = lane >> 4;
#pragma unroll
  for (int c = 0; c < 4; ++c) {
#pragma unroll
    for (int r = 0; r < 2; ++r) {
      const int K = 4 * c + r + 2 * hi;
      const float ang = -(TWO_PI_F / 16.0f) * (float)((M * K) & 15);
      float s, co;
      __sincosf(ang, &s, &co);
      F.fr[c][r] = co;
      F.fi[c][r] = s;
      F.fin[c][r] = -s;
    }
  }
}

// 16x16x16 complex matmul  C = F * Bmat  (conj(F) if INV).
// Bmat element (k,n) read at bre/bim[(k*16+n)*stride]: matches the ISA
// B-operand layout (lane L supplies K = 4c+r+2*(L/16), N = L%16).
template <bool CPLX, bool INV>
__device__ __forceinline__ void mm16(const F16c& F, const float* bre,
                                     const float* bim, int stride, int lane,
                                     v8f& cr, v8f& ci) {
  const int n = lane & 15;
  const int hi = lane >> 4;
#pragma unroll
  for (int c = 0; c < 4; ++c) {
    const int k0 = 4 * c + 2 * hi;
    v2f br;
    br[0] = bre[((k0 + 0) * 16 + n) * stride];
    br[1] = bre[((k0 + 1) * 16 + n) * stride];
    cr = WMMA_F32X4(F.fr[c], br, cr);
    ci = WMMA_F32X4(INV ? F.fin[c] : F.fi[c], br, ci);
    if (CPLX) {
      v2f bi;
      bi[0] = bim[((k0 + 0) * 16 + n) * stride];
      bi[1] = bim[((k0 + 1) * 16 + n) * stride];
      cr = WMMA_F32X4(INV ? F.fi[c] : F.fin[c], bi, cr);
      ci = WMMA_F32X4(F.fr[c], bi, ci);
    }
  }
}

// One wave = one 256-point complex FFT (four-step, radix-16):
//   stage1: S = F16 x reshape(input,16,16);  twiddle+transpose to staging;
//   stage2: R = F16 x W;  C-layout element (M,N) is X[16M+N].
// Same-wave DS ops are in-order so staging may alias the input.
template <bool INV, bool CPLX_IN>
__device__ __forceinline__ void fft256_wave(const F16c& F, int lane,
                                            const float* in_re,
                                            const float* in_im, int in_stride,
                                            float* s_re, float* s_im, v8f& rr,
                                            v8f& ri) {
  const int n = lane & 15;
  const int hi = lane >> 4;
  v8f cr = {}, ci = {};
  mm16<CPLX_IN, INV>(F, in_re, in_im, in_stride, lane, cr, ci);
  const float sgn = INV ? (TWO_PI_F / 256.0f) : -(TWO_PI_F / 256.0f);
#pragma unroll
  for (int v = 0; v < 8; ++v) {
    const int M = v + 8 * hi;  // k1
    const int N = n;           // a
    float ts, tc;
    __sincosf(sgn * (float)(M * N), &ts, &tc);
    const float wr = cr[v] * tc - ci[v] * ts;
    const float wi = cr[v] * ts + ci[v] * tc;
    s_re[N * 16 + M] = wr;  // transposed: stage-2 B-operand
    s_im[N * 16 + M] = wi;
  }
  asm volatile("" ::: "memory");  // keep stage-2 loads after the stores
  v8f r2 = {}, i2 = {};
  mm16<true, INV>(F, s_re, s_im, 1, lane, r2, i2);
  rr = r2;
  ri = i2;
}

// ---------------------------------------------------------------------------
// Pass 1: forward row FFTs (real input).  1 wave = 1 row (b,h).
// ws[b][h][kw] complex float2; stays RT-cached (pass 2/3 re-read it from L2).
// ---------------------------------------------------------------------------
__global__ __launch_bounds__(256) void fft_rows_fwd(
    const float* __restrict__ x, float2* __restrict__ ws) {
  __shared__ __align__(16) float stg[8 * 512];
  const int wave = threadIdx.x >> 5;
  const int lane = threadIdx.x & 31;
  const int b = blockIdx.x >> 5;
  const int h = ((blockIdx.x & 31) << 3) + wave;

  F16c F;
  build_F16(F, lane);
  float* s_re = &stg[wave * 512];
  float* s_im = s_re + 256;

  const v4f* row4 = (const v4f*)(x + ((size_t)b * 256 + h) * 256);
  v4f a0 = row4[lane * 2 + 0];
  v4f a1 = row4[lane * 2 + 1];
  *(v4f*)&s_re[lane * 8 + 0] = a0;
  *(v4f*)&s_re[lane * 8 + 4] = a1;

  v8f rr, ri;
  fft256_wave<false, false>(F, lane, s_re, nullptr, 1, s_re, s_im, rr, ri);

  float2* orow = ws + ((size_t)b * 256 + h) * 256;
  const int n = lane & 15;
  const int hi = lane >> 4;
#pragma unroll
  for (int v = 0; v < 8; ++v) {
    const int idx = 16 * (v + 8 * hi) + n;
    orow[idx] = make_float2(rr[v], ri[v]);
  }
}

// ---------------------------------------------------------------------------
// Pass 2: per column (b,kw): forward col FFT -> mask -> inverse col FFT,
// in place in ws.  Block = 8 waves = 8 adjacent columns.  The 256x8 complex
// tile is moved with async-to/from-LDS B128 DMA (ASYNCcnt), 64B per thread;
// the DFT-matrix sincos build overlaps the in-flight copy.
// Tile layout (interleaved): (kh, j) -> tile[kh*16 + 2j] (re), +1 (im).
// Mask: m = assigned ? pimp[(b+B/2)%B][cid[(kh+128)%256][(kw+128)%256]] : 1
// ---------------------------------------------------------------------------
__global__ __launch_bounds__(256) void fft_cols_mask(
    float2* __restrict__ ws, const float* __restrict__ pimp,
    const int* __restrict__ cid, int B) {
  __shared__ __align__(16) float tile[256 * 16];
  __shared__ __align__(16) float stg[8 * 512];
  const int wave = threadIdx.x >> 5;
  const int lane = threadIdx.x & 31;
  const int b = blockIdx.x >> 5;
  const int kw0 = (blockIdx.x & 31) << 3;

  // Async tile load: thread t DMAs ws[b][h=t][kw0..kw0+7] (64B) into its row.
  const unsigned lds_row = lds_addr_of(&tile[threadIdx.x * 16]);
  const unsigned gofs = (unsigned)(threadIdx.x * 2048 + kw0 * 8);
  const unsigned long long sbase =
      (unsigned long long)(uintptr_t)(ws + (size_t)b * 65536);
  asm volatile(
      "global_load_async_to_lds_b128 %0, %1, %2 offset:0\n\t"
      "global_load_async_to_lds_b128 %0, %1, %2 offset:16\n\t"
      "global_load_async_to_lds_b128 %0, %1, %2 offset:32\n\t"
      "global_load_async_to_lds_b128 %0, %1, %2 offset:48" ::"v"(lds_row),
      "v"(gofs), "s"(sbase)
      : "memory");

  F16c F;
  build_F16(F, lane);  // overlaps the async copy

  asm volatile("s_wait_asynccnt 0x0" ::: "memory");
  __syncthreads();

  float* s_re = &stg[wave * 512];
  float* s_im = s_re + 256;
  const int kw = kw0 + wave;

  // forward column FFT straight out of the interleaved tile (stride 16 floats)
  v8f cr, ci;
  fft256_wave<false, true>(F, lane, tile + 2 * wave, tile + 2 * wave + 1, 16,
                           s_re, s_im, cr, ci);

  // mask multiply; masked spectrum laid linearly into staging for the inverse
  const int n = lane & 15;
  const int hi = lane >> 4;
  int b2 = b + (B >> 1);
  if (b2 >= B) b2 -= B;
  const int wp = (kw + 128) & 255;
#pragma unroll
  for (int v = 0; v < 8; ++v) {
    const int kh = 16 * (v + 8 * hi) + n;
    const int hp = (kh + 128) & 255;
    const int cc = cid[hp * 256 + wp];
    const float m = (cc < 64) ? pimp[b2 * 64 + cc] : 1.0f;
    s_re[kh] = cr[v] * m;
    s_im[kh] = ci[v] * m;
  }
  asm volatile("" ::: "memory");

  v8f rr, ri;
  fft256_wave<true, true>(F, lane, s_re, s_im, 1, s_re, s_im, rr, ri);

#pragma unroll
  for (int v = 0; v < 8; ++v) {  // own column back to the interleaved tile
    const int kh = 16 * (v + 8 * hi) + n;
    *(float2*)&tile[kh * 16 + 2 * wave] = make_float2(rr[v], ri[v]);
  }
  // DS results must be in LDS before the async engine reads them.
  asm volatile("s_wait_dscnt 0x0" ::: "memory");
  __syncthreads();

  asm volatile(
      "global_store_async_from_lds_b128 %0, %1, %2 offset:0\n\t"
      "global_store_async_from_lds_b128 %0, %1, %2 offset:16\n\t"
      "global_store_async_from_lds_b128 %0, %1, %2 offset:32\n\t"
      "global_store_async_from_lds_b128 %0, %1, %2 offset:48" ::"v"(gofs),
      "v"(lds_row), "s"(sbase)
      : "memory");
  asm volatile("s_wait_asynccnt 0x0" ::: "memory");
}

// ---------------------------------------------------------------------------
// Pass 3: inverse row FFTs + magnitude + emit [x, ifft, |ifft-x|].
// ws reads are last-use and out stores are write-once: both non-temporal so
// they don't rinse the workspace out of L2.  2D inverse scale 1/65536 folded
// into the magnitude.
// ---------------------------------------------------------------------------
__global__ __launch_bounds__(256) void ifft_rows_out(
    const float2* __restrict__ ws, const float* __restrict__ x,
    float* __restrict__ out) {
  __shared__ __align__(16) float stg[8 * 512];
  const int wave = threadIdx.x >> 5;
  const int lane = threadIdx.x & 31;
  const int b = blockIdx.x >> 5;
  const int h = ((blockIdx.x & 31) << 3) + wave;

  F16c F;
  build_F16(F, lane);
  float* s_re = &stg[wave * 512];
  float* s_im = s_re + 256;

  const v4f* row4 = (const v4f*)ws + (size_t)(b * 256 + h) * 128;
#pragma unroll
  for (int i = 0; i < 4; ++i) {  // 64B per lane, NT (last use of ws)
    v4f q = __builtin_nontemporal_load(row4 + lane * 4 + i);
    const int e = lane * 8 + i * 2;
    s_re[e + 0] = q[0];
    s_im[e + 0] = q[1];
    s_re[e + 1] = q[2];
    s_im[e + 1] = q[3];
  }

  v8f rr, ri;
  fft256_wave<true, true>(F, lane, s_re, s_im, 1, s_re, s_im, rr, ri);

  const float scale = 1.0f / 65536.0f;
  const int n = lane & 15;
  const int hi = lane >> 4;
  const size_t base_in = ((size_t)b * 256 + h) * 256;
  const size_t base0 = ((size_t)b * 3 + 0) * 65536 + (size_t)h * 256;
  const size_t base1 = ((size_t)b * 3 + 1) * 65536 + (size_t)h * 256;
  const size_t base2 = ((size_t)b * 3 + 2) * 65536 + (size_t)h * 256;
#pragma unroll
  for (int v = 0; v < 8; ++v) {
    const int w = 16 * (v + 8 * hi) + n;
    const float xv = x[base_in + w];
    const float mag = sqrtf(rr[v] * rr[v] + ri[v] * ri[v]) * scale;
    __builtin_nontemporal_store(xv, &out[base0 + w]);
    __builtin_nontemporal_store(mag, &out[base1 + w]);
    __builtin_nontemporal_store(fabsf(mag - xv), &out[base2 + w]);
  }
}

// ---------------------------------------------------------------------------
extern "C" void kernel_launch(void* const* d_in, const int* in_sizes, int n_in,
                              void* d_out, int out_size, void* d_ws,
                              size_t ws_size, hipStream_t stream) {
  (void)n_in;
  (void)out_size;
  (void)ws_size;  // requires >= B*256*256*8 bytes (64 MB for B=128)

  const float* x = (const float*)d_in[0];
  const float* pimp = (const float*)d_in[1];
  const int* cid = (const int*)d_in[2];
  const int B = in_sizes[0] / (256 * 256);

  float2* ws = (float2*)d_ws;
  float* out = (float*)d_out;

  const int nblocks = B * 32;  // 8 rows/cols per 256-thread block
  fft_rows_fwd<<<nblocks, 256, 0, stream>>>(x, ws);
  fft_cols_mask<<<nblocks, 256, 0, stream>>>(ws, pimp, cid, B);
  ifft_rows_out<<<nblocks, 256, 0, stream>>>(ws, x, out);

  // Second tuple output: pattern_importance passthrough, appended after x_new.
  hipMemcpyAsync(out + (size_t)B * 3 * 65536, pimp,
                 (size_t)in_sizes[1] * sizeof(float),
                 hipMemcpyDeviceToDevice, stream);
}